// CapsuleLayer_19619410608187
// MI455X (gfx1250) — compile-verified
//
#include <hip/hip_runtime.h>
#include <cstdint>

// ---------------------------------------------------------------------------
// CapsuleLayer forward for MI455X (gfx1250, wave32).
// One workgroup (256 threads = 8 wave32) per batch element. All per-batch
// state lives in LDS; BE[b] is staged by the Tensor Data Mover in a single
// 2-D descriptor (with LDS pad for bank spreading). All matmuls use
// V_WMMA_F32_16X16X4_F32 (full fp32 fidelity). All inner-loop LDS loads are
// unconditional (zero-padded data instead of exec-mask predication).
// ---------------------------------------------------------------------------

typedef float v2f  __attribute__((ext_vector_type(2)));
typedef float v8f  __attribute__((ext_vector_type(8)));
typedef unsigned int u32x4 __attribute__((ext_vector_type(4)));
typedef int   i32x4 __attribute__((ext_vector_type(4)));
typedef int   i32x8 __attribute__((ext_vector_type(8)));

#define RT_ITERS 3
constexpr int L_  = 200;
constexpr int D_  = 64;
constexpr int O_  = 64;
constexpr int K_  = 8;
constexpr int LP  = 208;     // L padded to 13*16
constexpr int MSTR  = 68;    // LDS row stride (floats) for mapping
constexpr int BESTR = 68;    // LDS row stride for BE (set via TDM pad)
constexpr int SSTR  = 68;    // LDS row stride for S
constexpr int WSTR  = 16;    // W padded to 16 capsule columns (8..15 zero)
constexpr int GSTR  = 16;    // logits padded to 16 columns (8..15 junk)

// LDS carve-up (float offsets). BE (phase 1 only) overlays W/ZP/CAPS
// (routing phase only) in one union region.
constexpr int OFF_MAP  = 0;
constexpr int OFF_UNI  = OFF_MAP + LP * MSTR;       // union region
constexpr int UNI_SZ   = LP * BESTR;                // 14144 floats
constexpr int OFF_BE   = OFF_UNI;                   // phase 1
constexpr int OFF_W    = OFF_UNI;                   // routing: 208*16 = 3328
constexpr int OFF_ZP   = OFF_W + LP * WSTR;         // routing: 2*16*64 = 2048
constexpr int OFF_CAPS = OFF_ZP + 2 * 16 * O_;      // routing: 8*64 = 512
constexpr int OFF_S    = OFF_UNI + UNI_SZ;
constexpr int OFF_LG   = OFF_S + D_ * SSTR;
constexpr int SMEM_FLOATS = OFF_LG + LP * GSTR;

__device__ __forceinline__ v8f wmma4(v2f a, v2f b, v8f c) {
  // D = A(16x4 f32) * B(4x16 f32) + C(16x16 f32)
  return __builtin_amdgcn_wmma_f32_16x16x4_f32(
      /*neg_a=*/false, a, /*neg_b=*/false, b,
      /*c_mod=*/(short)0, c, /*reuse_a=*/false, /*reuse_b=*/false);
}

// One 2-D TDM load: global (dim1 rows x dim0 f32, row stride stride0 elems)
// -> LDS at lds_byte, padding pad_amt+1 dwords after every 2^(pad_int+1)
// dwords (=> padded LDS row stride).
__device__ __forceinline__ void tdm_load_2d_f32(
    uint32_t lds_byte, const void* gptr,
    uint32_t dim0, uint32_t dim1, uint32_t tile0, uint32_t tile1,
    uint32_t stride0, uint32_t pad_int, uint32_t pad_amt) {
  const uint64_t ga = (uint64_t)(uintptr_t)gptr;
  u32x4 g0;
  g0.x = 1u;                                    // count=1, user descriptor
  g0.y = lds_byte;                              // lds_addr [63:32]
  g0.z = (uint32_t)ga;                          // global_addr lo
  g0.w = (uint32_t)(ga >> 32) | (2u << 30);     // global_addr hi | type=2
  i32x8 g1;
  g1[0] = (int)((2u << 16)                      // data_size = 4B
                | (1u << 20)                    // pad_enable
                | (pad_int << 22) | (pad_amt << 25));
  g1[1] = (int)((dim0 & 0xFFFFu) << 16);        // tensor_dim0[15:0]
  g1[2] = (int)(((dim0 >> 16) & 0xFFFFu) | ((dim1 & 0xFFFFu) << 16));
  g1[3] = (int)(((dim1 >> 16) & 0xFFFFu) | ((tile0 & 0xFFFFu) << 16));
  g1[4] = (int)(tile1 & 0xFFFFu);               // tile_dim1, tile_dim2=0
  g1[5] = (int)stride0;                         // tensor_dim0_stride lo32
  g1[6] = 0;
  g1[7] = 0;
  i32x4 z4 = {0, 0, 0, 0};
#if __has_include(<hip/amd_detail/amd_gfx1250_TDM.h>)
  i32x8 z8 = {0, 0, 0, 0, 0, 0, 0, 0};
  __builtin_amdgcn_tensor_load_to_lds(g0, g1, z4, z4, z8, 0);
#else
  __builtin_amdgcn_tensor_load_to_lds(g0, g1, z4, z4, 0);
#endif
}

__global__ __launch_bounds__(256) void capsule_route_kernel(
    const float* __restrict__ be, const int* __restrict__ seqlen,
    const float* __restrict__ rl, const float* __restrict__ S,
    float* __restrict__ out) {
  extern __shared__ float sm[];
  const int b    = blockIdx.x;
  const int tid  = threadIdx.x;
  const int lane = tid & 31;
  const int wave = tid >> 5;
  const int lo16 = lane & 15;
  const int hi   = lane >> 4;   // 0: lanes 0-15, 1: lanes 16-31

  float* MAP  = sm + OFF_MAP;
  float* BE   = sm + OFF_BE;
  float* Ssh  = sm + OFF_S;
  float* LG   = sm + OFF_LG;
  float* W    = sm + OFF_W;
  float* ZP   = sm + OFF_ZP;
  float* CAPS = sm + OFF_CAPS;

  // ---- stage BE[b] (200x64 f32) into LDS rows 0..199 via TDM ------------
  if (wave == 0) {
    tdm_load_2d_f32((uint32_t)(uintptr_t)BE, be + (size_t)b * L_ * D_,
                    /*dim0=*/D_, /*dim1=*/L_, /*tile0=*/D_, /*tile1=*/L_,
                    /*stride0=*/D_, /*pad_int=*/5u /*64 dw*/, /*pad_amt=*/3u /*4 dw*/);
    __builtin_amdgcn_s_wait_tensorcnt(0);
  }
  // ---- zero BE pad rows 200..207 (disjoint from TDM-written region) -----
  for (int i = tid; i < (LP - L_) * BESTR; i += 256)
    BE[L_ * BESTR + i] = 0.0f;
  // ---- stage S (64x64) -> LDS -------------------------------------------
  for (int i = tid; i < D_ * O_; i += 256) {
    const int r = i >> 6, c = i & 63;
    Ssh[r * SSTR + c] = S[i];
  }
  // ---- init logits, padded to 16 cols (cols 8..15 junk, never read) -----
  for (int i = tid; i < LP * GSTR; i += 256) {
    const int r = i >> 4, k = i & 15;
    LG[i] = (r < L_ && k < K_) ? rl[((size_t)b * L_ + r) * K_ + k] : 0.0f;
  }
  const int slen = seqlen[b];
  __syncthreads();

  // ---- phase 1: MAP(LPx64) = BE(LPx64) @ S(64x64), WMMA f32 16x16x4 -----
  for (int t = wave; t < (LP / 16) * (O_ / 16); t += 8) {
    const int mt = t >> 2, nt = t & 3;
    const int row = mt * 16 + lo16;
    v8f acc = {};
#pragma unroll
    for (int kk = 0; kk < D_ / 4; ++kk) {
      const int k0 = kk * 4 + 2 * hi;
      v2f a, bb;
      a.x  = BE[row * BESTR + k0];
      a.y  = BE[row * BESTR + k0 + 1];
      bb.x = Ssh[k0 * SSTR + nt * 16 + lo16];
      bb.y = Ssh[(k0 + 1) * SSTR + nt * 16 + lo16];
      acc = wmma4(a, bb, acc);
    }
#pragma unroll
    for (int r = 0; r < 8; ++r)
      MAP[(mt * 16 + r + 8 * hi) * MSTR + nt * 16 + lo16] = acc[r];
  }
  __syncthreads();  // BE dead from here; union region becomes W/ZP/CAPS

  // ---- dynamic routing ---------------------------------------------------
  for (int it = 0; it < RT_ITERS; ++it) {
    // (a) weight = softmax_K(logits) * mask; cols 8..15 written as zero
    if (tid < LP) {
      float v[8], m = -1e30f;
#pragma unroll
      for (int k = 0; k < 8; ++k) { v[k] = LG[tid * GSTR + k]; m = fmaxf(m, v[k]); }
      float s = 0.0f;
#pragma unroll
      for (int k = 0; k < 8; ++k) { v[k] = __expf(v[k] - m); s += v[k]; }
      const float inv = ((tid < slen) ? 1.0f : 0.0f) / s;
#pragma unroll
      for (int k = 0; k < 8; ++k) W[tid * WSTR + k] = v[k] * inv;
#pragma unroll
      for (int k = 8; k < 16; ++k) W[tid * WSTR + k] = 0.0f;
    }
    __syncthreads();

    // (b) Z = W^T(16xLP) @ MAP(LPx64); 8 waves = 4 N-tiles x 2 L-halves
    {
      const int nt = wave & 3, half = wave >> 2;
      const int base = half * (LP / 2);
      v8f acc = {};
#pragma unroll 2
      for (int kk = 0; kk < (LP / 2) / 4; ++kk) {
        const int l0 = base + kk * 4 + 2 * hi;
        v2f a, bb;
        a.x  = W[l0 * WSTR + lo16];
        a.y  = W[(l0 + 1) * WSTR + lo16];
        bb.x = MAP[l0 * MSTR + nt * 16 + lo16];
        bb.y = MAP[(l0 + 1) * MSTR + nt * 16 + lo16];
        acc = wmma4(a, bb, acc);
      }
#pragma unroll
      for (int r = 0; r < 8; ++r)
        ZP[(half * 16 + r + 8 * hi) * O_ + nt * 16 + lo16] = acc[r];
    }
    __syncthreads();

    // (c) combine halves + squash; wave k handles capsule k
    {
      const int k = wave;  // 8 waves == 8 capsules
      const float z0 = ZP[k * O_ + lane] + ZP[(16 + k) * O_ + lane];
      const float z1 = ZP[k * O_ + lane + 32] + ZP[(16 + k) * O_ + lane + 32];
      float sq = z0 * z0 + z1 * z1;
#pragma unroll
      for (int off = 16; off > 0; off >>= 1) sq += __shfl_xor(sq, off, 32);
      const float factor = sq / (1.0f + sq) * rsqrtf(sq + 1e-9f);
      if (it == RT_ITERS - 1) {
        float* o = out + ((size_t)b * K_ + k) * O_;
        o[lane]      = factor * z0;
        o[lane + 32] = factor * z1;
      } else {
        CAPS[k * O_ + lane]      = factor * z0;
        CAPS[k * O_ + lane + 32] = factor * z1;
      }
    }
    if (it == RT_ITERS - 1) break;
    __syncthreads();

    // (d) logits += MAP(LPx64) @ CAPS^T(64x16); junk cols 8..15 written but
    //     never read (LG padded to 16 columns)
    for (int mt = wave; mt < LP / 16; mt += 8) {
      const int row = mt * 16 + lo16;
      const int kc = lo16 & 7;  // clamp junk columns into CAPS bounds
      v8f acc = {};
#pragma unroll
      for (int kk = 0; kk < O_ / 4; ++kk) {
        const int o0 = kk * 4 + 2 * hi;
        v2f a, bb;
        a.x  = MAP[row * MSTR + o0];
        a.y  = MAP[row * MSTR + o0 + 1];
        bb.x = CAPS[kc * O_ + o0];
        bb.y = CAPS[kc * O_ + o0 + 1];
        acc = wmma4(a, bb, acc);
      }
#pragma unroll
      for (int r = 0; r < 8; ++r)
        LG[(mt * 16 + r + 8 * hi) * GSTR + lo16] += acc[r];
    }
    __syncthreads();
  }
}

extern "C" void kernel_launch(void* const* d_in, const int* in_sizes, int n_in,
                              void* d_out, int out_size, void* d_ws, size_t ws_size,
                              hipStream_t stream) {
  (void)n_in; (void)out_size; (void)d_ws; (void)ws_size;
  const float* be = (const float*)d_in[0];   // [B,L,D] f32
  const int*   sl = (const int*)d_in[1];     // [B] i32
  const float* rl = (const float*)d_in[2];   // [B,L,K,1] f32
  const float* S  = (const float*)d_in[3];   // [D,O] f32
  float* out = (float*)d_out;                // [B,K,O] f32

  const int B = in_sizes[1];                 // batch count
  const size_t smem = (size_t)SMEM_FLOATS * sizeof(float);
  hipFuncSetAttribute((const void*)capsule_route_kernel,
                      hipFuncAttributeMaxDynamicSharedMemorySize, (int)smem);
  capsule_route_kernel<<<B, 256, smem, stream>>>(be, sl, rl, S, out);
}